// Axial_PFCU_Continuous_26302379720992
// MI455X (gfx1250) — compile-verified
//
#include <hip/hip_runtime.h>
#include <math.h>

typedef __attribute__((ext_vector_type(2))) float v2f;
typedef __attribute__((ext_vector_type(8))) float v8f;

#define Bn 8
#define Cn 96
#define Hn 128
#define Wn 128
#define HW (Hn * Wn)

// ---------------- LDS layout for k1 (floats) ----------------
#define WF_PITCH 97                     // 96 cols padded to 97 -> conflict-free
#define WPK_PITCH 17                    // [wc, 8 x H-tap w, 8 x W-tap w]
#define OFF_WF    0
#define OFF_WPK   (96 * WF_PITCH)       // 9312, size 96*17 = 1632
#define OFF_DGWH  (OFF_WPK + 96 * WPK_PITCH)   // 10944
#define OFF_DGWW  (OFF_DGWH + 288)
#define OFF_BNFS  (OFF_DGWW + 288)
#define OFF_BNFH  (OFF_BNFS + 96)
#define OFF_DGS   (OFF_BNFH + 96)
#define OFF_DGH   (OFF_DGS + 96)
#define OFF_ACTA  (OFF_DGH + 96)
#define LDS_TOTAL (OFF_ACTA + 96)       // 12000 floats = 48000 B

// off-center taps: ti 0..3 -> medium conv, 4..7 -> large conv; i = {0,1,3,4}
__device__ __constant__ const int TAP_I[8] = {0, 1, 3, 4, 0, 1, 3, 4};

// Per-wave tap geometry. H-side values are wave-uniform (h is readfirstlane'd)
// -> SGPRs; W-side values are per-lane.
struct TapGeom {
    int   rowPos0[8], rowPos1[8];   // clamped row * Wn   (uniform)
    float hW0[8],     hW1[8];       // mask-folded interp weights (uniform)
    int   colPos0[8], colPos1[8];   // h*Wn + clamped col (per-lane)
    float wW0[8],     wW1[8];       // (per-lane)
    int   hRow;                     // h*Wn (uniform)
};

template<bool FRAC>
__device__ __forceinline__ void k1_compute_acc(
    const float* __restrict__ xb, const float* __restrict__ lds,
    const TapGeom& tg, unsigned wlane, int p, int hi, v8f acc[6])
{
    for (int kk = 0; kk < 24; ++kk) {
        const int ch0 = kk * 4 + 2 * hi;   // this lane-half's K columns
        v2f bv;
#pragma unroll
        for (int j = 0; j < 2; ++j) {
            const int c = ch0 + j;
            const unsigned baseC = (unsigned)c * HW;
            const unsigned bw = baseC + wlane;

            // ---- batched sample loads (no branches, no waits in between) ----
            const float rc = xb[bw + (unsigned)tg.hRow];
            float rH0[8], rW0[8];
#pragma unroll
            for (int ti = 0; ti < 8; ++ti)
                rH0[ti] = xb[bw + (unsigned)tg.rowPos0[ti]];
#pragma unroll
            for (int ti = 0; ti < 8; ++ti)
                rW0[ti] = xb[baseC + (unsigned)tg.colPos0[ti]];
            float rH1[8], rW1[8];
            if (FRAC) {
#pragma unroll
                for (int ti = 0; ti < 8; ++ti)
                    rH1[ti] = xb[bw + (unsigned)tg.rowPos1[ti]];
#pragma unroll
                for (int ti = 0; ti < 8; ++ti)
                    rW1[ti] = xb[baseC + (unsigned)tg.colPos1[ti]];
            }

            // ---- arithmetic ----
            const float* wp = &lds[OFF_WPK + c * WPK_PITCH];
            float t = wp[0] * rc;   // wc = 2 + whm[2]+wwm[2]+whl[2]+wwl[2]
#pragma unroll
            for (int ti = 0; ti < 8; ++ti) {
                float s = tg.hW0[ti] * rH0[ti];
                if (FRAC) s = fmaf(tg.hW1[ti], rH1[ti], s);
                t = fmaf(wp[1 + ti], s, t);
            }
#pragma unroll
            for (int ti = 0; ti < 8; ++ti) {
                float s = tg.wW0[ti] * rW0[ti];
                if (FRAC) s = fmaf(tg.wW1[ti], rW1[ti], s);
                t = fmaf(wp[9 + ti], s, t);
            }
            if (j == 0) bv.x = t; else bv.y = t;
        }

        // ---- preload all A tiles, then back-to-back WMMAs ----
        v2f av[6];
#pragma unroll
        for (int mt = 0; mt < 6; ++mt) {
            const int arow = OFF_WF + (mt * 16 + p) * WF_PITCH + ch0;
            av[mt].x = lds[arow];
            av[mt].y = lds[arow + 1];
        }
#pragma unroll
        for (int mt = 0; mt < 6; ++mt)
            acc[mt] = __builtin_amdgcn_wmma_f32_16x16x4_f32(
                false, av[mt], false, bv, (short)0, acc[mt], false, false);
    }
}

// ============================================================
// K1: fused axial-DW (m+l) -> pointwise 96x96 (WMMA f32) -> BN
//     + edges/anchor BN -> PReLU, writes "out_pre" into d_out.
// One wave = 16 pixels (strip along W). blockDim = 128 (4 waves).
// grid = B*H*(W/16)/4 = 2048.
// ============================================================
__global__ __launch_bounds__(128) void k1_fused(
    const float* __restrict__ x,
    const float* __restrict__ rm_p, const float* __restrict__ whm_g, const float* __restrict__ wwm_g,
    const float* __restrict__ rl_p, const float* __restrict__ whl_g, const float* __restrict__ wwl_g,
    const float* __restrict__ wfuse,
    const float* __restrict__ bnf_g, const float* __restrict__ bnf_b,
    const float* __restrict__ bnf_m, const float* __restrict__ bnf_v,
    const float* __restrict__ dgwh_g, const float* __restrict__ dgww_g,
    const float* __restrict__ dg_g, const float* __restrict__ dg_b,
    const float* __restrict__ dg_m, const float* __restrict__ dg_v,
    const float* __restrict__ acta_g,
    float* __restrict__ out)
{
    __shared__ float lds[LDS_TOTAL];

    // ---- stage weights / params into LDS ----
    for (int i = threadIdx.x; i < Cn * Cn; i += 128)
        lds[OFF_WF + (i / Cn) * WF_PITCH + (i % Cn)] = wfuse[i];
    for (int i = threadIdx.x; i < Cn * 3; i += 128) {
        lds[OFF_DGWH + i] = dgwh_g[i];
        lds[OFF_DGWW + i] = dgww_g[i];
    }
    if (threadIdx.x < Cn) {
        const int c = threadIdx.x;
        // packed per-channel tap weights
        float* wp = &lds[OFF_WPK + c * WPK_PITCH];
        wp[0] = 2.0f + whm_g[c * 5 + 2] + wwm_g[c * 5 + 2]
                     + whl_g[c * 5 + 2] + wwl_g[c * 5 + 2];
#pragma unroll
        for (int ti = 0; ti < 8; ++ti) {
            const int i = TAP_I[ti];
            wp[1 + ti] = (ti < 4) ? whm_g[c * 5 + i] : whl_g[c * 5 + i];
            wp[9 + ti] = (ti < 4) ? wwm_g[c * 5 + i] : wwl_g[c * 5 + i];
        }
        // BN / PReLU params
        const float s1 = bnf_g[c] * rsqrtf(bnf_v[c] + 1e-5f);
        lds[OFF_BNFS + c] = s1;
        lds[OFF_BNFH + c] = bnf_b[c] - bnf_m[c] * s1;
        const float s2 = dg_g[c] * rsqrtf(dg_v[c] + 1e-5f);
        lds[OFF_DGS + c] = s2;
        lds[OFF_DGH + c] = dg_b[c] - dg_m[c] * s2;
        lds[OFF_ACTA + c] = acta_g[c];
    }
    __syncthreads();

    const int lane = threadIdx.x & 31;
    const int p    = lane & 15;   // pixel within strip (N index)
    const int hi   = lane >> 4;   // lane half (K split)

    // strip index is wave-uniform: force it scalar so b/h/wt live in SGPRs
    const int wave  = __builtin_amdgcn_readfirstlane((int)(threadIdx.x >> 5));
    const int strip = blockIdx.x * 4 + wave;
    const int wt = strip & 7;
    const int h  = (strip >> 3) & (Hn - 1);
    const int b  = strip >> 10;
    const int w  = wt * 16 + p;   // per-lane

    const float* xb = x + (size_t)b * Cn * HW;   // uniform base -> saddr loads

    // Broadcast dilation scalars so frac-dispatch is provably wave-uniform.
    const float rm = __uint_as_float(
        __builtin_amdgcn_readfirstlane(__float_as_uint(fmaxf(rm_p[0], 1.0f))));
    const float rl = __uint_as_float(
        __builtin_amdgcn_readfirstlane(__float_as_uint(fmaxf(rl_p[0], 1.0f))));

    // Build channel-invariant tap geometry (clamped offsets + masked weights).
    TapGeom tg;
    tg.hRow = h * Wn;
    bool anyfrac = false;
#pragma unroll
    for (int ti = 0; ti < 8; ++ti) {
        const float r  = (ti < 4) ? rm : rl;
        const int   k  = TAP_I[ti] - 2;       // {-2,-1,1,2}
        const float of = (float)k * r;
        const float ff = floorf(of);
        const int   fi = (int)ff;
        const float q  = of - ff;
        anyfrac = anyfrac || (q > 0.0f);

        const int h0 = h + fi;                                   // uniform
        tg.rowPos0[ti] = min(max(h0, 0), Hn - 1) * Wn;
        tg.hW0[ti]     = ((unsigned)h0 < (unsigned)Hn) ? (1.0f - q) : 0.0f;
        tg.rowPos1[ti] = min(max(h0 + 1, 0), Hn - 1) * Wn;
        tg.hW1[ti]     = ((unsigned)(h0 + 1) < (unsigned)Hn) ? q : 0.0f;

        const int w0 = w + fi;                                   // per-lane
        tg.colPos0[ti] = h * Wn + min(max(w0, 0), Wn - 1);
        tg.wW0[ti]     = ((unsigned)w0 < (unsigned)Wn) ? (1.0f - q) : 0.0f;
        tg.colPos1[ti] = h * Wn + min(max(w0 + 1, 0), Wn - 1);
        tg.wW1[ti]     = ((unsigned)(w0 + 1) < (unsigned)Wn) ? q : 0.0f;
    }

    v8f acc[6] = {};   // 6 M-tiles x 8 VGPRs f32 accumulators
    if (anyfrac) k1_compute_acc<true >(xb, lds, tg, (unsigned)w, p, hi, acc);
    else         k1_compute_acc<false>(xb, lds, tg, (unsigned)w, p, hi, acc);

    // ---- epilogue: BN(fused) + anchor + PReLU (branchless) ----
    const int   hm  = (h > 0)      ? h - 1 : 0;       // uniform
    const int   hp  = (h < Hn - 1) ? h + 1 : h;
    const float mh0 = (h > 0)      ? 1.0f : 0.0f;
    const float mh2 = (h < Hn - 1) ? 1.0f : 0.0f;
    const int   wm  = (w > 0)      ? w - 1 : 0;       // per-lane
    const int   wp_ = (w < Wn - 1) ? w + 1 : w;
    const float mw0 = (w > 0)      ? 1.0f : 0.0f;
    const float mw2 = (w < Wn - 1) ? 1.0f : 0.0f;

#pragma unroll
    for (int mt = 0; mt < 6; ++mt) {
#pragma unroll
        for (int v = 0; v < 8; ++v) {
            const int o = mt * 16 + v + 8 * hi;   // C/D layout: lanes16-31 -> M+8
            const unsigned baseO = (unsigned)o * HW;
            const float fused = acc[mt][v] * lds[OFF_BNFS + o] + lds[OFF_BNFH + o];
            const float xc  = xb[baseO + (unsigned)(h * Wn + w)];
            const float xn  = xb[baseO + (unsigned)(hm * Wn + w)];
            const float xs  = xb[baseO + (unsigned)(hp * Wn + w)];
            const float xw_ = xb[baseO + (unsigned)(h * Wn + wm)];
            const float xe  = xb[baseO + (unsigned)(h * Wn + wp_)];
            float e = (lds[OFF_DGWH + o * 3 + 1] + lds[OFF_DGWW + o * 3 + 1]) * xc;
            e = fmaf(lds[OFF_DGWH + o * 3 + 0] * mh0, xn, e);
            e = fmaf(lds[OFF_DGWH + o * 3 + 2] * mh2, xs, e);
            e = fmaf(lds[OFF_DGWW + o * 3 + 0] * mw0, xw_, e);
            e = fmaf(lds[OFF_DGWW + o * 3 + 2] * mw2, xe, e);
            const float anch = (xc + e) * lds[OFF_DGS + o] + lds[OFF_DGH + o];
            const float pre = fused + anch;
            const float ov = pre >= 0.f ? pre : lds[OFF_ACTA + o] * pre;
            out[((size_t)(b * Cn + o) * Hn + h) * Wn + w] = ov;
        }
    }
}

// ============================================================
// K2: per-(b,c) plane -> row means (over W) and col means (over H)
// grid = B*C = 768, block = 128 (4 waves; wave handles 32 rows).
// b128 loads: one float4 per lane covers a full 128-wide row per wave.
// ============================================================
__global__ __launch_bounds__(128) void k2_means(
    const float* __restrict__ dout,
    float* __restrict__ rowmean, float* __restrict__ colmean)
{
    __shared__ float colpart[4 * 128];
    const float4* pb4 = (const float4*)(dout + (size_t)blockIdx.x * HW);
    const int wave = threadIdx.x >> 5;
    const int lane = threadIdx.x & 31;

    float c0 = 0.f, c1 = 0.f, c2 = 0.f, c3 = 0.f;  // cols 4*lane .. 4*lane+3
    for (int i = 0; i < 32; ++i) {
        const int h = wave * 32 + i;
        const float4 v = pb4[h * (Wn / 4) + lane];
        c0 += v.x; c1 += v.y; c2 += v.z; c3 += v.w;
        float rsum = (v.x + v.y) + (v.z + v.w);
        for (int off = 16; off > 0; off >>= 1)
            rsum += __shfl_xor(rsum, off, 32);
        if (lane == 0)
            rowmean[(size_t)blockIdx.x * Hn + h] = rsum * (1.0f / (float)Wn);
    }
    colpart[wave * 128 + 4 * lane + 0] = c0;
    colpart[wave * 128 + 4 * lane + 1] = c1;
    colpart[wave * 128 + 4 * lane + 2] = c2;
    colpart[wave * 128 + 4 * lane + 3] = c3;
    __syncthreads();
    {
        const int t = threadIdx.x;
        float s = colpart[t] + colpart[128 + t] + colpart[256 + t] + colpart[384 + t];
        colmean[(size_t)blockIdx.x * Wn + t] = s * (1.0f / (float)Hn);
    }
}

// ============================================================
// K3: coordinate-attention MLP per (b, pos), pos in [0, H+W)
// grid = B*(H+W) = 2048, block = 96
// ============================================================
__global__ __launch_bounds__(96) void k3_att(
    const float* __restrict__ rowmean, const float* __restrict__ colmean,
    const float* __restrict__ w1,
    const float* __restrict__ g, const float* __restrict__ bb,
    const float* __restrict__ mm, const float* __restrict__ vv,
    const float* __restrict__ aa,
    const float* __restrict__ wh, const float* __restrict__ ww,
    float* __restrict__ a_h, float* __restrict__ a_w)
{
    __shared__ float mv[96];
    __shared__ float yv[8];
    const int b   = blockIdx.x >> 8;   // H+W = 256
    const int pos = blockIdx.x & 255;
    const int c   = threadIdx.x;
    const bool isH = (pos < Hn);

    mv[c] = isH ? rowmean[((size_t)b * Cn + c) * Hn + pos]
                : colmean[((size_t)b * Cn + c) * Wn + (pos - Hn)];
    __syncthreads();

    if (c < 8) {
        float s = 0.f;
        for (int j = 0; j < Cn; ++j) s += w1[c * Cn + j] * mv[j];
        const float sc = g[c] * rsqrtf(vv[c] + 1e-5f);
        const float bn = (s - mm[c]) * sc + bb[c];
        yv[c] = bn >= 0.f ? bn : aa[c] * bn;
    }
    __syncthreads();

    const float* w2 = isH ? wh : ww;
    float s2 = 0.f;
#pragma unroll
    for (int j = 0; j < 8; ++j) s2 += w2[c * 8 + j] * yv[j];
    const float a = 1.0f / (1.0f + expf(-s2));
    if (isH) a_h[((size_t)b * Cn + c) * Hn + pos] = a;
    else     a_w[((size_t)b * Cn + c) * Wn + (pos - Hn)] = a;
}

// ============================================================
// K4: in-place gating: out *= a_h[b,c,h] * a_w[b,c,w]
// b128 load/store: one float4 (4 consecutive w) per thread.
// ============================================================
__global__ __launch_bounds__(256) void k4_scale(
    float* __restrict__ out,
    const float* __restrict__ a_h, const float* __restrict__ a_w)
{
    const size_t q = (size_t)blockIdx.x * 256 + threadIdx.x;   // quad index
    const size_t idx = q * 4;
    const int w0 = (int)(idx & (Wn - 1));          // aligned to 4
    const int h  = (int)((idx >> 7) & (Hn - 1));
    const size_t plane = idx >> 14;                // b*C + c

    float4 v  = *(float4*)(out + idx);
    const float4 aw = *(const float4*)(a_w + plane * Wn + w0);
    const float ah = a_h[plane * Hn + h];
    v.x = v.x * ah * aw.x;
    v.y = v.y * ah * aw.y;
    v.z = v.z * ah * aw.z;
    v.w = v.w * ah * aw.w;
    *(float4*)(out + idx) = v;
}

// ============================================================
extern "C" void kernel_launch(void* const* d_in, const int* in_sizes, int n_in,
                              void* d_out, int out_size, void* d_ws, size_t ws_size,
                              hipStream_t stream)
{
    const float* x     = (const float*)d_in[0];
    const float* r_m   = (const float*)d_in[1];
    const float* wh_m  = (const float*)d_in[2];
    const float* ww_m  = (const float*)d_in[3];
    const float* r_l   = (const float*)d_in[4];
    const float* wh_l  = (const float*)d_in[5];
    const float* ww_l  = (const float*)d_in[6];
    const float* wfuse = (const float*)d_in[7];
    const float* bnf_g = (const float*)d_in[8];
    const float* bnf_b = (const float*)d_in[9];
    const float* bnf_m = (const float*)d_in[10];
    const float* bnf_v = (const float*)d_in[11];
    const float* dg_wh = (const float*)d_in[12];
    const float* dg_ww = (const float*)d_in[13];
    const float* dg_g  = (const float*)d_in[14];
    const float* dg_b  = (const float*)d_in[15];
    const float* dg_m  = (const float*)d_in[16];
    const float* dg_v  = (const float*)d_in[17];
    const float* act_a = (const float*)d_in[18];
    const float* ca_w1 = (const float*)d_in[19];
    const float* ca_g  = (const float*)d_in[20];
    const float* ca_b  = (const float*)d_in[21];
    const float* ca_m  = (const float*)d_in[22];
    const float* ca_v  = (const float*)d_in[23];
    const float* ca_a  = (const float*)d_in[24];
    const float* ca_wh = (const float*)d_in[25];
    const float* ca_ww = (const float*)d_in[26];

    float* out = (float*)d_out;
    float* ws  = (float*)d_ws;

    const size_t NBC = (size_t)Bn * Cn;        // 768
    float* rowmean = ws;                       // 768*128
    float* colmean = ws + NBC * Hn;            // 768*128
    float* a_h     = colmean + NBC * Wn;       // 768*128
    float* a_w     = a_h + NBC * Hn;           // 768*128

    // K1: 8192 strips / 4 waves per block
    k1_fused<<<dim3(2048), dim3(128), 0, stream>>>(
        x, r_m, wh_m, ww_m, r_l, wh_l, ww_l, wfuse,
        bnf_g, bnf_b, bnf_m, bnf_v, dg_wh, dg_ww,
        dg_g, dg_b, dg_m, dg_v, act_a, out);

    k2_means<<<dim3((unsigned)NBC), dim3(128), 0, stream>>>(out, rowmean, colmean);

    k3_att<<<dim3(Bn * (Hn + Wn)), dim3(96), 0, stream>>>(
        rowmean, colmean, ca_w1, ca_g, ca_b, ca_m, ca_v, ca_a, ca_wh, ca_ww,
        a_h, a_w);

    const unsigned totalQuads = (Bn * Cn * Hn * Wn) / 4;   // 3,145,728
    k4_scale<<<dim3(totalQuads / 256), dim3(256), 0, stream>>>(out, a_h, a_w);
}